// HashEncoder_249108103673
// MI455X (gfx1250) — compile-verified
//
#include <hip/hip_runtime.h>
#include <math.h>

#define NLVL 16
#define BLK  256
#define OSTR 34   // padded LDS row stride in dwords (32 data + 2 pad) -> conflict-free ds_store_b64

// Levels >= DENSE_LVLS have res >= 111 (robust vs pow() rounding), so res^3 > 2^19
// and the table size is exactly T_MAX = 2^19 (power of two) per the reference config.
#define DENSE_LVLS 6
#define TMASK ((1u << 19) - 1u)

typedef float v2f __attribute__((ext_vector_type(2)));

struct HashParams {
  const float* x;
  float* out;
  const float* tab[NLVL];
  unsigned long long M[NLVL];   // floor(2^64/T)+1 : exact u32 mod magic (dense levels)
  unsigned T[NLVL];
  unsigned P1m[NLVL];           // 2654435761 % T
  unsigned P2m[NLVL];           // 805459861  % T
  float    resf[NLVL];
  int      resm1[NLVL];
  unsigned n;
};

__device__ __forceinline__ unsigned umodT(unsigned s, unsigned T, unsigned long long M) {
  unsigned q = (unsigned)(((__uint128_t)s * (__uint128_t)M) >> 64);
  return s - q * T;
}

__global__ __launch_bounds__(BLK) void hashgrid_enc(HashParams p) {
  __shared__ float shx[BLK * 3];      // 3 KB staged x
  __shared__ float sho[BLK * OSTR];   // 34 KB padded output tile
  const unsigned tid = threadIdx.x;
  const unsigned pt0 = blockIdx.x * BLK;
  const unsigned totalf = p.n * 3u;

  // ---- async-stage this block's 256 x-triples into LDS (ASYNCcnt path) ----
#pragma unroll
  for (int j = 0; j < 3; ++j) {
    const unsigned fi = pt0 * 3u + (unsigned)(tid + BLK * j);
    if (fi < totalf) {
      const unsigned ldsb = (unsigned)(size_t)&shx[tid + BLK * j];          // LDS byte offset
      const unsigned long long ga = (unsigned long long)(size_t)(p.x + fi); // 64-bit global addr
      asm volatile("global_load_async_to_lds_b32 %0, %1, off"
                   :: "v"(ldsb), "v"(ga) : "memory");
    }
  }
  // prefetch the x stream two blocks ahead (speculative; OOB silently dropped)
  __builtin_prefetch(p.x + (size_t)pt0 * 3u + tid + 6u * BLK, 0, 3);
  asm volatile("s_wait_asynccnt 0" ::: "memory");
  __syncthreads();

  const float px = shx[tid * 3 + 0];
  const float py = shx[tid * 3 + 1];
  const float pz = shx[tid * 3 + 2];

#pragma unroll
  for (int l = 0; l < NLVL; ++l) {
    const bool POW2 = (l >= DENSE_LVLS);                 // compile-time after unroll
    const float rf = p.resf[l];
    const int   rm1 = p.resm1[l];
    const unsigned T  = POW2 ? (TMASK + 1u) : p.T[l];
    const unsigned Py = POW2 ? (2654435761u & TMASK) : p.P1m[l];
    const unsigned Pz = POW2 ? (805459861u  & TMASK) : p.P2m[l];
    const float* __restrict__ tab = p.tab[l];

    const float xs0 = px * rf, xs1 = py * rf, xs2 = pz * rf;
    int a0 = (int)xs0, a1 = (int)xs1, a2 = (int)xs2;
    a0 = a0 < 0 ? 0 : (a0 > rm1 ? rm1 : a0);
    a1 = a1 < 0 ? 0 : (a1 > rm1 ? rm1 : a1);
    a2 = a2 < 0 ? 0 : (a2 > rm1 ? rm1 : a2);
    const float w1x = xs0 - (float)a0, w1y = xs1 - (float)a1, w1z = xs2 - (float)a2;
    const float w0x = 1.0f - w1x, w0y = 1.0f - w1y, w0z = 1.0f - w1z;
    const int b0 = (a0 + 1 > rm1) ? rm1 : a0 + 1;
    const int b1 = (a1 + 1 > rm1) ? rm1 : a1 + 1;
    const int b2 = (a2 + 1 > rm1) ? rm1 : a2 + 1;

    const unsigned hx[2] = { (unsigned)a0,      (unsigned)b0 };
    const unsigned hy[2] = { (unsigned)a1 * Py, (unsigned)b1 * Py };
    const unsigned hz[2] = { (unsigned)a2 * Pz, (unsigned)b2 * Pz };
    const float    wz[2] = { w0z, w1z };
    // factored corner weights: 4 muls + 8 muls
    const float wxy[4] = { w0x * w0y, w0x * w1y, w1x * w0y, w1x * w1y };

    float f0 = 0.0f, f1 = 0.0f;
#pragma unroll
    for (int k = 0; k < 8; ++k) {                 // k = (dx,dy,dz), matches reference offsets
      const int dx = (k >> 2) & 1, dy = (k >> 1) & 1, dz = k & 1;
      const unsigned s = hx[dx] + hy[dy] + hz[dz];
      const unsigned idx = POW2 ? (s & TMASK) : umodT(s, T, p.M[l]);
      const v2f v = *(const v2f*)(tab + 2u * idx); // global_load_b64, SGPR table base (RT: stay in L2)
      const float w = wxy[2 * dx + dy] * wz[dz];
      f0 = __builtin_fmaf(w, v.x, f0);
      f1 = __builtin_fmaf(w, v.y, f1);
    }
    v2f r; r.x = f0; r.y = f1;
    *(v2f*)(&sho[tid * OSTR + 2 * l]) = r;        // conflict-free ds_store_b64 (stride 34)
  }
  __syncthreads();

  // ---- coalesced, non-temporal writeback: don't evict L2-resident tables ----
  const unsigned outbase = pt0 * (2u * NLVL);
  const unsigned totalo  = p.n * (2u * NLVL);
  if (outbase + BLK * 2u * NLVL <= totalo) {
    v2f* __restrict__ o2 = (v2f*)(p.out + outbase);
#pragma unroll
    for (int k = 0; k < NLVL; ++k) {              // 16 x 256 lanes = 4096 v2f = 32 KB
      const unsigned i2 = tid + BLK * (unsigned)k;
      const unsigned t = i2 >> 4, j2 = i2 & 15u;
      const v2f v = *(const v2f*)(&sho[t * OSTR + 2u * j2]);
      __builtin_nontemporal_store(v, &o2[i2]);    // global_store_b64 TH=NT, fully coalesced
    }
  } else {
    for (unsigned i = tid; i < BLK * 2u * NLVL; i += BLK) {
      const unsigned gi = outbase + i;
      if (gi < totalo) p.out[gi] = sho[(i >> 5) * OSTR + (i & 31u)];
    }
  }
}

extern "C" void kernel_launch(void* const* d_in, const int* in_sizes, int n_in,
                              void* d_out, int out_size, void* d_ws, size_t ws_size,
                              hipStream_t stream) {
  (void)n_in; (void)out_size; (void)d_ws; (void)ws_size;
  HashParams p;
  p.x   = (const float*)d_in[0];
  p.out = (float*)d_out;
  p.n   = (unsigned)(in_sizes[0] / 3);
  const double growth = pow(2048.0 / 16.0, 1.0 / 15.0);  // same libm pow as Python
  for (int l = 0; l < NLVL; ++l) {
    p.tab[l] = (const float*)d_in[1 + l];
    const unsigned T = (unsigned)(in_sizes[1 + l] / 2);  // table entries (F=2)
    p.T[l]   = T;
    p.M[l]   = ~0ull / T + 1ull;
    p.P1m[l] = (unsigned)(2654435761ull % T);
    p.P2m[l] = (unsigned)(805459861ull  % T);
    const int res = (int)(16.0 * pow(growth, (double)l));
    p.resf[l]  = (float)res;
    p.resm1[l] = res - 1;
  }
  dim3 grid((p.n + BLK - 1) / BLK), block(BLK);
  hipLaunchKernelGGL(hashgrid_enc, grid, block, 0, stream, p);
}